// FeaturePropagation_2688649527509
// MI455X (gfx1250) — compile-verified
//
#include <hip/hip_runtime.h>
#include <stdint.h>

// Feature propagation on MI455X (gfx1250).
// Memory-bound SpMM-style gather/accumulate; working set ~130MB -> resident in 192MB L2.
// CSR built on device each call; iterations use the CDNA5 async global->LDS gather
// pipeline (global_load_async_to_lds_b128 + s_wait_asynccnt, ASYNCcnt-tracked).

#define TPB 256
#define WPB 8          // waves per block in prop kernel (wave32)
#define N_ITER 20

// ---------------------------------------------------------------- propagation iteration
// One wave32 per output row; lane l owns float4 slice [4l..4l+3] of the 128 features.
// Neighbor rows gathered via async global->LDS loads, double-buffered 4 rows at a time.
// Chunk metadata (indices + weights) is prefetched one chunk ahead so the only stall
// at s_wait_asynccnt is true async completion.

__device__ __forceinline__ void load_meta(const int* __restrict__ colS,
                                          const float* __restrict__ wS,
                                          int s, int cnt, int ch,
                                          int (&c)[4], float (&w)[4])
{
#pragma unroll
    for (int k = 0; k < 4; ++k) {
        int j = ch * 4 + k;
        bool v = (j < cnt);
        c[k] = v ? colS[s + j] : 0;     // pad with row 0
        w[k] = v ? wS[s + j] : 0.0f;    // ...and weight 0
    }
}

__device__ __forceinline__ void issue_chunk(const float4* __restrict__ src,
                                            const int (&c)[4],
                                            const float4* stageLane, int lane)
{
#pragma unroll
    for (int k = 0; k < 4; ++k) {
        const float4* gp = src + (size_t)c[k] * 32 + lane;   // 16B/lane, 512B/row
        unsigned int lp = (unsigned int)(uintptr_t)(stageLane + (size_t)k * 32); // LDS byte addr
        asm volatile("global_load_async_to_lds_b128 %0, %1, off"
                     :: "v"(lp), "v"(gp) : "memory");        // ASYNCcnt++
    }
}

__global__ __launch_bounds__(WPB * 32) void prop_kernel(
    const float4* __restrict__ src, float4* __restrict__ dst,
    const int* __restrict__ rowStart, const int* __restrict__ rowCnt,
    const int* __restrict__ colS, const float* __restrict__ wS,
    const unsigned char* __restrict__ mask, int N)
{
    __shared__ float4 stage[WPB][2][4][32];   // 32KB: per-wave double buffer, 4 rows x 512B
    const int wave = threadIdx.x >> 5;
    const int lane = threadIdx.x & 31;
    const int r = blockIdx.x * WPB + wave;
    if (r >= N) return;
    if (mask[r]) return;                      // dst row already holds x (init), never touched
    const int s   = rowStart[r];
    const int cnt = rowCnt[r];
    float4 acc = make_float4(0.f, 0.f, 0.f, 0.f);
    if (cnt > 0) {
        const int nch = (cnt + 3) >> 2;
        int cCur[4]; float wCur[4];
        load_meta(colS, wS, s, cnt, 0, cCur, wCur);
        issue_chunk(src, cCur, &stage[wave][0][0][lane], lane);
        for (int ch = 0; ch < nch; ++ch) {
            int cNxt[4] = {0, 0, 0, 0};
            float wNxt[4] = {0.f, 0.f, 0.f, 0.f};
            if (ch + 1 < nch) {
                load_meta(colS, wS, s, cnt, ch + 1, cNxt, wNxt);
                issue_chunk(src, cNxt, &stage[wave][(ch + 1) & 1][0][lane], lane);
                // 8 outstanding; async loads complete in order => <=4 means chunk ch is in LDS
                asm volatile("s_wait_asynccnt 4" ::: "memory");
            } else {
                asm volatile("s_wait_asynccnt 0" ::: "memory");
            }
            const int b = ch & 1;
#pragma unroll
            for (int k = 0; k < 4; ++k) {
                float4 v = stage[wave][b][k][lane];          // ds_load_b128
                acc.x = fmaf(wCur[k], v.x, acc.x);
                acc.y = fmaf(wCur[k], v.y, acc.y);
                acc.z = fmaf(wCur[k], v.z, acc.z);
                acc.w = fmaf(wCur[k], v.w, acc.w);
            }
#pragma unroll
            for (int k = 0; k < 4; ++k) { cCur[k] = cNxt[k]; wCur[k] = wNxt[k]; }
        }
    }
    dst[(size_t)r * 32 + lane] = acc;
}

// ---------------------------------------------------------------- histogram
__global__ void deg_hist_kernel(const int* __restrict__ ei,
                                const unsigned char* __restrict__ mask,
                                float* __restrict__ deg,
                                int* __restrict__ rowCnt, int E)
{
    int e = blockIdx.x * blockDim.x + threadIdx.x;
    if (e >= E) return;
    int r = ei[e];          // row
    int c = ei[E + e];      // col
    unsafeAtomicAdd(&deg[c], 1.0f);         // native global_atomic_add_f32
    if (!mask[r]) atomicAdd(&rowCnt[r], 1); // drop edges whose output row is masked
}

__global__ void rsqrt_kernel(float* __restrict__ deg, int N)
{
    int i = blockIdx.x * blockDim.x + threadIdx.x;
    if (i >= N) return;
    float d = deg[i];
    deg[i] = (d > 0.0f) ? rsqrtf(d) : 0.0f;
}

// ---------------------------------------------------------------- 2-level exclusive scan
__global__ void scan_pass1(const int* __restrict__ cnt, int* __restrict__ blockSum, int N)
{
    __shared__ int sm[1024];
    int i = blockIdx.x * 1024 + threadIdx.x;
    sm[threadIdx.x] = (i < N) ? cnt[i] : 0;
    __syncthreads();
    for (int o = 512; o > 0; o >>= 1) {
        if (threadIdx.x < o) sm[threadIdx.x] += sm[threadIdx.x + o];
        __syncthreads();
    }
    if (threadIdx.x == 0) blockSum[blockIdx.x] = sm[0];
}

__global__ void scan_pass2(int* __restrict__ blockSum, int nB)
{
    if (threadIdx.x == 0 && blockIdx.x == 0) {
        int run = 0;
        for (int i = 0; i < nB; ++i) { int t = blockSum[i]; blockSum[i] = run; run += t; }
    }
}

__global__ void scan_pass3(const int* __restrict__ cnt, const int* __restrict__ blockOff,
                           int* __restrict__ rowStart, int N)
{
    __shared__ int sm[1024];
    int i = blockIdx.x * 1024 + threadIdx.x;
    int v = (i < N) ? cnt[i] : 0;
    sm[threadIdx.x] = v;
    __syncthreads();
    for (int o = 1; o < 1024; o <<= 1) {
        int t = (threadIdx.x >= o) ? sm[threadIdx.x - o] : 0;
        __syncthreads();
        sm[threadIdx.x] += t;
        __syncthreads();
    }
    if (i < N) rowStart[i] = blockOff[blockIdx.x] + sm[threadIdx.x] - v; // exclusive
}

// ---------------------------------------------------------------- CSR bucket fill
__global__ void bucket_kernel(const int* __restrict__ ei,
                              const unsigned char* __restrict__ mask,
                              const float* __restrict__ dis,
                              const int* __restrict__ rowStart,
                              int* __restrict__ cursor,
                              int* __restrict__ colS, float* __restrict__ wS, int E)
{
    int e = blockIdx.x * blockDim.x + threadIdx.x;
    if (e >= E) return;
    int r = ei[e];
    if (mask[r]) return;                    // masked rows never need their edges
    int c = ei[E + e];
    int pos = rowStart[r] + atomicAdd(&cursor[r], 1);
    colS[pos] = c;
    wS[pos]   = dis[r] * dis[c];
}

// ---------------------------------------------------------------- init: out0 and masked rows
__global__ void init_kernel(const float4* __restrict__ x,
                            const unsigned char* __restrict__ mask,
                            float4* __restrict__ out, float4* __restrict__ buf, int nChunks)
{
    int i = blockIdx.x * blockDim.x + threadIdx.x;
    if (i >= nChunks) return;
    int n = i >> 5;                         // D=128 -> 32 float4 per row
    if (mask[n]) { float4 v = x[i]; out[i] = v; buf[i] = v; }  // fixed forever in both buffers
    else          { out[i] = make_float4(0.f, 0.f, 0.f, 0.f); }
}

// ---------------------------------------------------------------- launch
extern "C" void kernel_launch(void* const* d_in, const int* in_sizes, int n_in,
                              void* d_out, int out_size, void* d_ws, size_t ws_size,
                              hipStream_t stream)
{
    const float*         x    = (const float*)d_in[0];
    const int*           ei   = (const int*)d_in[1];
    const unsigned char* mask = (const unsigned char*)d_in[2]; // jnp.bool_ = 1 byte
    const int N = in_sizes[2];
    const int E = in_sizes[1] / 2;
    // D == 128 fixed by the reference.

    char* ws = (char*)d_ws;
    size_t off = 0;
    auto alloc = [&](size_t bytes) -> void* {
        void* p = (void*)(ws + off);
        off += (bytes + 255) & ~(size_t)255;
        return p;
    };
    float* deg      = (float*)alloc((size_t)N * 4);   // becomes deg_inv_sqrt in place
    int*   rowCnt   = (int*)  alloc((size_t)N * 4);
    int*   rowStart = (int*)  alloc((size_t)N * 4);
    int*   cursor   = (int*)  alloc((size_t)N * 4);
    int*   blockSum = (int*)  alloc(4096);
    int*   colS     = (int*)  alloc((size_t)E * 4);
    float* wS       = (float*)alloc((size_t)E * 4);
    float4* buf     = (float4*)alloc((size_t)N * 128 * 4);
    (void)ws_size; (void)n_in; (void)out_size;

    hipMemsetAsync(deg,    0, (size_t)N * 4, stream);
    hipMemsetAsync(rowCnt, 0, (size_t)N * 4, stream);
    hipMemsetAsync(cursor, 0, (size_t)N * 4, stream);

    const int gE = (E + TPB - 1) / TPB;
    const int gN = (N + TPB - 1) / TPB;
    deg_hist_kernel<<<gE, TPB, 0, stream>>>(ei, mask, deg, rowCnt, E);
    rsqrt_kernel<<<gN, TPB, 0, stream>>>(deg, N);

    const int nB = (N + 1023) / 1024;
    scan_pass1<<<nB, 1024, 0, stream>>>(rowCnt, blockSum, N);
    scan_pass2<<<1, 1, 0, stream>>>(blockSum, nB);
    scan_pass3<<<nB, 1024, 0, stream>>>(rowCnt, blockSum, rowStart, N);

    bucket_kernel<<<gE, TPB, 0, stream>>>(ei, mask, deg, rowStart, cursor, colS, wS, E);

    float4* out4 = (float4*)d_out;
    const float4* x4 = (const float4*)x;
    const int nChunks = N * 32;
    init_kernel<<<(nChunks + TPB - 1) / TPB, TPB, 0, stream>>>(x4, mask, out4, buf, nChunks);

    // Ping-pong: out0 lives in d_out; after 20 (even) iterations result is back in d_out.
    const int gP = (N + WPB - 1) / WPB;
    for (int it = 1; it <= N_ITER; ++it) {
        const float4* s4 = (it & 1) ? (const float4*)out4 : (const float4*)buf;
        float4*       d4 = (it & 1) ? buf : out4;
        prop_kernel<<<gP, WPB * 32, 0, stream>>>(s4, d4, rowStart, rowCnt, colS, wS, mask, N);
    }
}